// LSTM_network_19688130085781
// MI455X (gfx1250) — compile-verified
//
#include <hip/hip_runtime.h>
#include <math.h>

#define SEQ_LEN 524288
#define HID     7
#define NGATES  28   // 4*HID

typedef __attribute__((ext_vector_type(2))) float v2f;
typedef __attribute__((ext_vector_type(8))) float v8f;

// ---------- cross-lane helpers (wave32) ----------
static __device__ __forceinline__ float bcast(float v, int lane) {
  // uniform broadcast -> v_readlane_b32 (result in SGPR, cheap FMA operand)
  return __int_as_float(__builtin_amdgcn_readlane(__float_as_int(v), lane));
}
static __device__ __forceinline__ float gatherb(float v, int byteAddr) {
  // per-lane-varying gather -> ds_bpermute_b32 (index in bytes)
  return __int_as_float(__builtin_amdgcn_ds_bpermute(byteAddr, __float_as_int(v)));
}
static __device__ __forceinline__ float fast_tanh(float x) {
#if __has_builtin(__builtin_amdgcn_tanhf)
  return __builtin_amdgcn_tanhf(x);   // gfx1250 v_tanh_f32
#else
  return tanhf(x);
#endif
}

// =====================================================================
// Kernel 1: xg[t, 0:28] = x[t, 0:3] @ W_ih.T + (b_ih + b_hh)
// V_WMMA_F32_16X16X4_F32: A = 16 timesteps x K4 (K=3 zero-padded),
// B = K4 x 16 gate columns (two tiles cover 28 gates), bias folded into C.
// One wave handles 16 timesteps; 8 waves/block -> 128 timesteps/block.
// =====================================================================
__global__ __launch_bounds__(256) void xg_wmma_kernel(
    const float* __restrict__ x, const float* __restrict__ W_ih,
    const float* __restrict__ b_ih, const float* __restrict__ b_hh,
    float* __restrict__ xg) {
  const int lane = threadIdx.x & 31;
  const int wave = threadIdx.x >> 5;
  const int t0   = (blockIdx.x * 8 + wave) * 16;
  const int half = lane >> 4;      // 0: K={0,1}, 1: K={2,3}
  const int m    = lane & 15;
  const bool hiCol = (16 + m < NGATES);   // cols 28..31 are padding

  // A 16x4 fp32: lanes 0-15 hold (K0,K1) of row m, lanes 16-31 hold (K2,0)
  const int row = t0 + m;
  v2f a;
  if (half == 0) { a.x = x[row * 3 + 0]; a.y = x[row * 3 + 1]; }
  else           { a.x = x[row * 3 + 2]; a.y = 0.0f; }

  // B 4x16: B[k][n] = W_ih[n][k] (K=3 row zero; cols>=28 zero)
  v2f b0, b1;
  {
    const int n0 = m;
    if (half == 0) { b0.x = W_ih[n0 * 3 + 0]; b0.y = W_ih[n0 * 3 + 1]; }
    else           { b0.x = W_ih[n0 * 3 + 2]; b0.y = 0.0f; }
    const int n1 = 16 + m;
    if (hiCol) {
      if (half == 0) { b1.x = W_ih[n1 * 3 + 0]; b1.y = W_ih[n1 * 3 + 1]; }
      else           { b1.x = W_ih[n1 * 3 + 2]; b1.y = 0.0f; }
    } else { b1.x = 0.0f; b1.y = 0.0f; }
  }

  // C: bias depends only on column n -> same value in all 8 C regs
  const float bias0 = b_ih[m] + b_hh[m];
  const float bias1 = hiCol ? (b_ih[16 + m] + b_hh[16 + m]) : 0.0f;
  v8f c0, c1;
#pragma unroll
  for (int v = 0; v < 8; ++v) { c0[v] = bias0; c1[v] = bias1; }

  v8f d0 = __builtin_amdgcn_wmma_f32_16x16x4_f32(false, a, false, b0,
                                                 (short)0, c0, false, false);
  v8f d1 = __builtin_amdgcn_wmma_f32_16x16x4_f32(false, a, false, b1,
                                                 (short)0, c1, false, false);

  // D layout: element (m = v + 8*half, n = lane&15). Guard hoisted out of the
  // store loops so each loop is a straight-line store clause.
  float* base = xg + (size_t)(t0 + half * 8) * NGATES;
#pragma unroll
  for (int v = 0; v < 8; ++v) base[v * NGATES + m] = d0[v];
  if (hiCol) {
#pragma unroll
    for (int v = 0; v < 8; ++v) base[v * NGATES + 16 + m] = d1[v];
  }
}

// =====================================================================
// One LSTM step, one wave32. Lane j < 28 owns gate j:
//   lanes 0-6: i, 7-13: f, 14-20: g, 21-27: o.
// State h,c valid on lanes 0-6. sigmoid(x) = 0.5 + 0.5*tanh(0.5*x).
// adF/adG/adO are hoisted ds_bpermute byte addresses.
// =====================================================================
#define LSTM_STEP(XGC)                                                     \
  {                                                                        \
    float h0 = bcast(h, 0), h1 = bcast(h, 1), h2 = bcast(h, 2);            \
    float h3 = bcast(h, 3), h4 = bcast(h, 4), h5 = bcast(h, 5);            \
    float h6 = bcast(h, 6);                                                \
    float a0 = fmaf(w0, h0, (XGC));                                        \
    float a1 = fmaf(w1, h1, w2 * h2);                                      \
    float a2 = fmaf(w3, h3, w4 * h4);                                      \
    float a3 = fmaf(w5, h5, w6 * h6);                                      \
    float accv = (a0 + a1) + (a2 + a3);                                    \
    float th   = fast_tanh(accv * presc);                                  \
    float actv = is_g ? th : fmaf(0.5f, th, 0.5f);                         \
    float fv = gatherb(actv, adF);                                         \
    float gv = gatherb(actv, adG);                                         \
    float ov = gatherb(actv, adO);                                         \
    c = fmaf(fv, c, actv * gv); /* lanes 0-6: actv == sigmoid(i_j) */      \
    h = ov * fast_tanh(c);                                                 \
  }

// =====================================================================
// Kernel 2: sequential scan, single wave32. xg is L2-resident (58.7 MB
// < 192 MB L2); 4-step register pipeline on the loads (~240 cyc cover)
// plus an unconditional 32-step-ahead global_prefetch_b8 (L2 -> L0).
// h_t overwrites the consumed xg row (cols 0..6) in place; stores are
// batched into one predicated region per 4-step block.
// =====================================================================
__global__ void lstm_scan_kernel(const float* __restrict__ W_hh,
                                 float* __restrict__ xg) {
  const int  j        = threadIdx.x;           // 0..31, one wave
  const bool act_lane = (j < NGATES);
  float w0 = 0, w1 = 0, w2 = 0, w3 = 0, w4 = 0, w5 = 0, w6 = 0;
  if (act_lane) {
    const float* wr = W_hh + j * HID;
    w0 = wr[0]; w1 = wr[1]; w2 = wr[2]; w3 = wr[3];
    w4 = wr[4]; w5 = wr[5]; w6 = wr[6];
  }
  const bool  is_g  = (j >= 2 * HID) && (j < 3 * HID);
  const float presc = is_g ? 1.0f : 0.5f;
  const int   idx   = act_lane ? j : 0;
  const int   adF   = (j + HID) << 2;      // hoisted bpermute byte addrs
  const int   adG   = (j + 2 * HID) << 2;
  const int   adO   = (j + 3 * HID) << 2;

  float h = 0.0f, c = 0.0f;

  float cur0 = xg[0 * NGATES + idx];
  float cur1 = xg[1 * NGATES + idx];
  float cur2 = xg[2 * NGATES + idx];
  float cur3 = xg[3 * NGATES + idx];

  for (int t = 0; t < SEQ_LEN; t += 4) {
    const int tb = (t + 4 < SEQ_LEN) ? (t + 4) : t;   // clamped tail (unused)
    float n0 = xg[(tb + 0) * NGATES + idx];
    float n1 = xg[(tb + 1) * NGATES + idx];
    float n2 = xg[(tb + 2) * NGATES + idx];
    float n3 = xg[(tb + 3) * NGATES + idx];
    // all lanes prefetch their own element 32 rows ahead (no branch,
    // fire-and-forget global_prefetch, no counter traffic)
    __builtin_prefetch(xg + (size_t)((t + 32) & (SEQ_LEN - 1)) * NGATES + idx, 0, 1);

    LSTM_STEP(cur0); const float hs0 = h;
    LSTM_STEP(cur1); const float hs1 = h;
    LSTM_STEP(cur2); const float hs2 = h;
    LSTM_STEP(cur3); const float hs3 = h;

    if (j < HID) {                        // single exec toggle per block
      float* r = xg + (size_t)t * NGATES + j;
      r[0 * NGATES] = hs0;
      r[1 * NGATES] = hs1;
      r[2 * NGATES] = hs2;
      r[3 * NGATES] = hs3;
    }

    cur0 = n0; cur1 = n1; cur2 = n2; cur3 = n3;
  }
}

// =====================================================================
// Kernel 3: y = relu(h @ W1.T + b1) @ W2.T + b2; out = (y0, y1^2).
// Bandwidth-bound; weights are wave-uniform -> scalar loads.
// =====================================================================
__global__ __launch_bounds__(256) void head_kernel(
    const float* __restrict__ hs, int stride,
    const float* __restrict__ W1, const float* __restrict__ b1,
    const float* __restrict__ W2, const float* __restrict__ b2,
    float* __restrict__ out) {
  const int t = blockIdx.x * 256 + threadIdx.x;
  if (t >= SEQ_LEN) return;
  const float* hp = hs + (size_t)t * stride;
  float hv[HID];
#pragma unroll
  for (int k = 0; k < HID; ++k) hv[k] = hp[k];
  float z[HID];
#pragma unroll
  for (int r = 0; r < HID; ++r) {
    float a = b1[r];
#pragma unroll
    for (int k = 0; k < HID; ++k) a = fmaf(W1[r * HID + k], hv[k], a);
    z[r] = a > 0.0f ? a : 0.0f;
  }
  float y0 = b2[0], y1 = b2[1];
#pragma unroll
  for (int k = 0; k < HID; ++k) {
    y0 = fmaf(W2[k], z[k], y0);
    y1 = fmaf(W2[HID + k], z[k], y1);
  }
  out[t * 2 + 0] = y0;
  out[t * 2 + 1] = y1 * y1;
}

// =====================================================================
// Fallback: fully fused single-wave scan (xg + MLP head inline),
// zero workspace. Used only if ws_size is too small for xg.
// =====================================================================
__global__ void lstm_fused_kernel(
    const float* __restrict__ x,  const float* __restrict__ W_ih,
    const float* __restrict__ W_hh, const float* __restrict__ b_ih,
    const float* __restrict__ b_hh, const float* __restrict__ W1,
    const float* __restrict__ b1,   const float* __restrict__ W2,
    const float* __restrict__ b2,   float* __restrict__ out) {
  const int  j        = threadIdx.x;
  const bool act_lane = (j < NGATES);
  float w0 = 0, w1 = 0, w2 = 0, w3 = 0, w4 = 0, w5 = 0, w6 = 0;
  float u0 = 0, u1 = 0, u2 = 0, bj = 0;
  if (act_lane) {
    const float* wr = W_hh + j * HID;
    w0 = wr[0]; w1 = wr[1]; w2 = wr[2]; w3 = wr[3];
    w4 = wr[4]; w5 = wr[5]; w6 = wr[6];
    u0 = W_ih[j * 3 + 0]; u1 = W_ih[j * 3 + 1]; u2 = W_ih[j * 3 + 2];
    bj = b_ih[j] + b_hh[j];
  }
  float q0 = 0, q1 = 0, q2 = 0, q3 = 0, q4 = 0, q5 = 0, q6 = 0, c1j = 0;
  if (j < HID) {
    const float* qr = W1 + j * HID;
    q0 = qr[0]; q1 = qr[1]; q2 = qr[2]; q3 = qr[3];
    q4 = qr[4]; q5 = qr[5]; q6 = qr[6];
    c1j = b1[j];
  }
  const int rrow = (j < 2) ? j : 0;
  float r0 = W2[rrow * HID + 0], r1 = W2[rrow * HID + 1], r2 = W2[rrow * HID + 2];
  float r3 = W2[rrow * HID + 3], r4 = W2[rrow * HID + 4], r5 = W2[rrow * HID + 5];
  float r6 = W2[rrow * HID + 6], b2j = b2[rrow];

  const bool  is_g  = (j >= 2 * HID) && (j < 3 * HID);
  const float presc = is_g ? 1.0f : 0.5f;
  const int   lidx  = (j < 12) ? j : 0;   // lanes 0-11 carry 4 rows x 3 of x
  const int   adF   = (j + HID) << 2;
  const int   adG   = (j + 2 * HID) << 2;
  const int   adO   = (j + 3 * HID) << 2;

  float h = 0.0f, c = 0.0f;
  float xcur = x[lidx];                    // rows 0..3

  for (int t = 0; t < SEQ_LEN; t += 4) {
    const int tb = (t + 4 < SEQ_LEN) ? (t + 4) : t;
    float xnext = x[(size_t)tb * 3 + lidx];
    float ybuf[4];
#pragma unroll
    for (int s = 0; s < 4; ++s) {
      float x0 = bcast(xcur, 3 * s + 0);
      float x1 = bcast(xcur, 3 * s + 1);
      float x2 = bcast(xcur, 3 * s + 2);
      float xgc = fmaf(u0, x0, fmaf(u1, x1, fmaf(u2, x2, bj)));
      LSTM_STEP(xgc);
      // inline MLP head (off the recurrence critical path)
      float h0 = bcast(h, 0), h1 = bcast(h, 1), h2 = bcast(h, 2);
      float h3 = bcast(h, 3), h4 = bcast(h, 4), h5 = bcast(h, 5);
      float h6 = bcast(h, 6);
      float zz = fmaf(q0, h0, fmaf(q1, h1, fmaf(q2, h2,
                 fmaf(q3, h3, fmaf(q4, h4, fmaf(q5, h5, fmaf(q6, h6, c1j)))))));
      zz = zz > 0.0f ? zz : 0.0f;
      float z0 = bcast(zz, 0), z1 = bcast(zz, 1), z2 = bcast(zz, 2);
      float z3 = bcast(zz, 3), z4 = bcast(zz, 4), z5 = bcast(zz, 5);
      float z6 = bcast(zz, 6);
      float y = fmaf(r0, z0, fmaf(r1, z1, fmaf(r2, z2,
                fmaf(r3, z3, fmaf(r4, z4, fmaf(r5, z5, fmaf(r6, z6, b2j)))))));
      ybuf[s] = (j == 0) ? y : y * y;
    }
    if (j < 2) {                           // single exec toggle per block
      out[(t + 0) * 2 + j] = ybuf[0];
      out[(t + 1) * 2 + j] = ybuf[1];
      out[(t + 2) * 2 + j] = ybuf[2];
      out[(t + 3) * 2 + j] = ybuf[3];
    }
    xcur = xnext;
  }
}

extern "C" void kernel_launch(void* const* d_in, const int* in_sizes, int n_in,
                              void* d_out, int out_size, void* d_ws, size_t ws_size,
                              hipStream_t stream) {
  (void)in_sizes; (void)n_in; (void)out_size;
  const float* x    = (const float*)d_in[0];
  const float* W_ih = (const float*)d_in[1];
  const float* W_hh = (const float*)d_in[2];
  const float* b_ih = (const float*)d_in[3];
  const float* b_hh = (const float*)d_in[4];
  const float* W1   = (const float*)d_in[5];
  const float* b1   = (const float*)d_in[6];
  const float* W2   = (const float*)d_in[7];
  const float* b2   = (const float*)d_in[8];
  float* out = (float*)d_out;

  const size_t xg_bytes = (size_t)SEQ_LEN * NGATES * sizeof(float);  // 58.7 MB
  if (ws_size >= xg_bytes) {
    float* xg = (float*)d_ws;
    xg_wmma_kernel<<<SEQ_LEN / 128, 256, 0, stream>>>(x, W_ih, b_ih, b_hh, xg);
    lstm_scan_kernel<<<1, 32, 0, stream>>>(W_hh, xg);
    head_kernel<<<SEQ_LEN / 256, 256, 0, stream>>>(xg, NGATES, W1, b1, W2, b2, out);
  } else {
    lstm_fused_kernel<<<1, 32, 0, stream>>>(x, W_ih, W_hh, b_ih, b_hh,
                                            W1, b1, W2, b2, out);
  }
}